// CtcLossLayer_81277961109772
// MI455X (gfx1250) — compile-verified
//
#include <hip/hip_runtime.h>
#include <hip/hip_bf16.h>
#include <math.h>

typedef __attribute__((ext_vector_type(2))) float v2f;
typedef __attribute__((ext_vector_type(8))) float v8f;

#define NEGV (-1e30f)

// ---------------------------------------------------------------------------
// Kernel 1: per (b,t) row of 128 logits -> logsumexp (lse) and argmax (pred).
// One wave32 handles 16 rows. Sum of exp() computed with V_WMMA_F32_16X16X4_F32
// (full fp32): A = 16x4 chunk of exp values, B = ones (layout invariant),
// C/D accumulates over 32 chunks of 4 columns.
// A layout (ISA 7.12.2, 32-bit A 16x4): lanes 0-15 = M, VGPR0 holds K=0
// (lanes 0-15) / K=2 (lanes 16-31); VGPR1 holds K=1 / K=3.
// ---------------------------------------------------------------------------
__global__ __launch_bounds__(256) void k_row_stats(const float* __restrict__ logits,
                                                   float* __restrict__ lse,
                                                   int* __restrict__ pred,
                                                   int BT)
{
    const int lane = threadIdx.x & 31;
    const int wave = blockIdx.x * 8 + (threadIdx.x >> 5);
    const int row0 = wave * 16;
    if (row0 >= BT) return;

    const int m     = lane & 15;            // matrix row this lane feeds
    const int hi    = lane >> 4;            // 0: K%4 in {0,1}; 1: K%4 in {2,3}
    int row = row0 + m; if (row >= BT) row = BT - 1;   // tail clamp (keeps EXEC full)
    const float2* __restrict__ rp2 = (const float2*)(logits + (size_t)row * 128);

    // ---- pass 1: max + first-occurrence argmax over this lane's half ----
    float mx = -3.4e38f; int amx = 0;
    #pragma unroll 4
    for (int c = 0; c < 32; ++c) {
        const int k0 = 4 * c + 2 * hi;
        float2 v = rp2[2 * c + hi];
        if (v.x > mx) { mx = v.x; amx = k0; }
        if (v.y > mx) { mx = v.y; amx = k0 + 1; }
    }
    // combine the two halves of the row (lane ^ 16); ties -> smaller index
    {
        float omx = __shfl_xor(mx, 16);
        int   oam = __shfl_xor(amx, 16);
        if (omx > mx || (omx == mx && oam < amx)) { mx = omx; amx = oam; }
    }

    // ---- pass 2: exp + WMMA row-sum accumulation ----
    v8f acc = {};
    v2f ones; ones[0] = 1.0f; ones[1] = 1.0f;
    for (int c = 0; c < 32; ++c) {
        float2 v = rp2[2 * c + hi];
        v2f a; a[0] = expf(v.x - mx); a[1] = expf(v.y - mx);
        acc = __builtin_amdgcn_wmma_f32_16x16x4_f32(
            /*neg_a=*/false, a, /*neg_b=*/false, ones,
            /*c_mod=*/(short)0, acc, /*reuse_a=*/false, /*reuse_b=*/false);
    }

    // ---- extract: D VGPR r -> M=r (lanes 0-15) and M=r+8 (lanes 16-31) ----
    float rowsum = 0.0f;
    #pragma unroll
    for (int r = 0; r < 8; ++r) {
        float vlo = __shfl(acc[r], 0);     // row r
        float vhi = __shfl(acc[r], 16);    // row r + 8
        if (lane == r)     rowsum = vlo;
        if (lane == r + 8) rowsum = vhi;
    }
    if (lane < 16 && (row0 + lane) < BT) {
        const int r = row0 + lane;
        lse[r]  = mx + logf(rowsum);       // logsumexp over the 128 classes
        pred[r] = amx;                     // argmax class
    }
}

// ---------------------------------------------------------------------------
// Kernel 2: CTC forward DP. One block (128 threads) per example; alpha[S=2L+1]
// in LDS. The serial T-loop is software-pipelined: each thread issues one
// coalesced global_load_async_to_lds_b32 for logits row t+1 while computing
// step t from the LDS double buffer; s_wait_asynccnt + the workgroup barrier
// provide the handoff (rows complete in issue order).
// ---------------------------------------------------------------------------
__global__ __launch_bounds__(128) void k_ctc(const float* __restrict__ logits,
                                             const float* __restrict__ lse,
                                             const int* __restrict__ labels,
                                             const int* __restrict__ label_len,
                                             const int* __restrict__ logit_len,
                                             float* __restrict__ loss,
                                             int T, int C, int L)
{
    __shared__ float alpha[128];
    __shared__ float lrow[2][128];        // double-buffered logits row (C=128)
    const int b = blockIdx.x;
    const int s = threadIdx.x;
    const int S = 2 * L + 1;

    int ext = 0; bool allow = false;
    if (s < S) {
        ext = (s & 1) ? labels[b * L + (s >> 1)] : 0;
        int ext2 = (s >= 2) ? ((s & 1) ? labels[b * L + ((s - 2) >> 1)] : 0) : -1;
        allow = (ext != 0) && (ext != ext2);
    }
    const float* __restrict__ lg = logits + (size_t)b * T * C;
    const float* __restrict__ ls = lse + (size_t)b * T;
    const int tlen = logit_len[b];

    // prologue: async prefetch rows 0 and 1 (one dword per thread, coalesced)
    {
        unsigned d0 = (unsigned)(uintptr_t)&lrow[0][s];
        unsigned long long g0 = (unsigned long long)(uintptr_t)(lg + s);
        asm volatile("global_load_async_to_lds_b32 %0, %1, off"
                     :: "v"(d0), "v"(g0) : "memory");
        if (T > 1) {
            unsigned d1 = (unsigned)(uintptr_t)&lrow[1][s];
            unsigned long long g1 = (unsigned long long)(uintptr_t)(lg + C + s);
            asm volatile("global_load_async_to_lds_b32 %0, %1, off"
                         :: "v"(d1), "v"(g1) : "memory");
        }
    }

    // t = 0 init: need row 0 resident and visible to all waves
    asm volatile("s_wait_asynccnt 0x1" ::: "memory");
    __syncthreads();
    alpha[s] = (s < 2 && s < S) ? (lrow[0][ext] - ls[0]) : NEGV;
    __syncthreads();

    for (int t = 1; t < T; ++t) {
        float a0 = NEGV, a1 = NEGV, a2 = NEGV;
        if (s < S) {
            a0 = alpha[s];
            a1 = (s >= 1) ? alpha[s - 1] : NEGV;
            a2 = (s >= 2 && allow) ? alpha[s - 2] : NEGV;
        }
        // prefetch row t+1 into the buffer that held row t-1 (all reads of
        // row t-1 completed before last iteration's closing barrier)
        if (t + 1 < T) {
            unsigned dd = (unsigned)(uintptr_t)&lrow[(t + 1) & 1][s];
            unsigned long long gg =
                (unsigned long long)(uintptr_t)(lg + (size_t)(t + 1) * C + s);
            asm volatile("global_load_async_to_lds_b32 %0, %1, off"
                         :: "v"(dd), "v"(gg) : "memory");
        }
        asm volatile("s_wait_asynccnt 0x1" ::: "memory");   // row t landed
        __syncthreads();          // alpha reads done + row t visible block-wide
        if (s < S && t < tlen) {
            float mm = fmaxf(a0, fmaxf(a1, a2));
            float l3 = mm + logf(expf(a0 - mm) + expf(a1 - mm) + expf(a2 - mm));
            alpha[s] = l3 + (lrow[t & 1][ext] - ls[t]);
        }
        __syncthreads();          // alpha writes done
    }

    if (s == 0) {
        const int se = 2 * label_len[b];
        float a1v = alpha[se];
        float a2v = (se >= 1) ? alpha[se - 1] : NEGV;
        float mm  = fmaxf(a1v, a2v);
        loss[b] = -(mm + logf(expf(a1v - mm) + expf(a2v - mm)));
    }
}

// ---------------------------------------------------------------------------
// Kernel 3: fused greedy decode (merge repeats, drop blanks) + Levenshtein
// row-DP against labels. One thread per example; row has <= L+1 columns.
// ---------------------------------------------------------------------------
__global__ __launch_bounds__(256) void k_edit(const int* __restrict__ pred,
                                              const int* __restrict__ labels,
                                              const int* __restrict__ label_len,
                                              const int* __restrict__ logit_len,
                                              float* __restrict__ ed,
                                              int B, int T, int L)
{
    const int b = blockIdx.x * blockDim.x + threadIdx.x;
    if (b >= B) return;
    const int lb = label_len[b];
    const int tl = logit_len[b];
    const int* __restrict__ pr  = pred + (size_t)b * T;
    const int* __restrict__ lab = labels + (size_t)b * L;

    int row[65];                           // supports L <= 64
    for (int j = 0; j <= lb; ++j) row[j] = j;

    int prev = -1;
    for (int t = 0; t < T; ++t) {
        const int p = (t < tl) ? pr[t] : 0;
        const bool keep = (t < tl) && (p != 0) && (p != prev);
        prev = p;
        if (keep) {                        // one DP row with symbol p
            int diag = row[0];
            row[0] = row[0] + 1;
            for (int j = 1; j <= lb; ++j) {
                const int cost = (p != lab[j - 1]) ? 1 : 0;
                int nv = row[j] + 1;
                int lft = row[j - 1] + 1;
                if (lft < nv) nv = lft;
                int dg = diag + cost;
                if (dg < nv) nv = dg;
                diag = row[j];
                row[j] = nv;
            }
        }
    }
    ed[b] = (float)row[lb] / (float)((lb > 0) ? lb : 1);
}

// ---------------------------------------------------------------------------
// Kernel 4: deterministic fixed-tree mean reduction -> d_out[0] (loss mean),
// d_out[1] (edit-distance mean). Always fully (re)writes d_out.
// ---------------------------------------------------------------------------
__global__ __launch_bounds__(256) void k_reduce(const float* __restrict__ loss,
                                                const float* __restrict__ ed,
                                                float* __restrict__ out, int B)
{
    __shared__ float s1[256], s2[256];
    const int t = threadIdx.x;
    float a = 0.0f, c = 0.0f;
    for (int i = t; i < B; i += 256) { a += loss[i]; c += ed[i]; }
    s1[t] = a; s2[t] = c;
    __syncthreads();
    for (int st = 128; st > 0; st >>= 1) {
        if (t < st) { s1[t] += s1[t + st]; s2[t] += s2[t + st]; }
        __syncthreads();
    }
    if (t == 0) { out[0] = s1[0] / (float)B; out[1] = s2[0] / (float)B; }
}

// ---------------------------------------------------------------------------
extern "C" void kernel_launch(void* const* d_in, const int* in_sizes, int n_in,
                              void* d_out, int out_size, void* d_ws, size_t ws_size,
                              hipStream_t stream)
{
    const int*   labels    = (const int*)d_in[0];
    const float* logits    = (const float*)d_in[1];
    const int*   label_len = (const int*)d_in[2];
    const int*   logit_len = (const int*)d_in[3];

    const int B  = in_sizes[2];
    const int L  = in_sizes[0] / B;
    const int C  = 128;                       // fixed by this problem / kernel 1+2 layout
    const int T  = in_sizes[1] / (B * C);
    const int BT = B * T;

    // workspace layout: lse[BT] f32 | pred[BT] i32 | loss[B] f32 | ed[B] f32
    float* lse  = (float*)d_ws;
    int*   pred = (int*)(lse + BT);
    float* loss = (float*)(pred + BT);
    float* ed   = loss + B;

    const int waves   = (BT + 15) / 16;
    const int blocks1 = (waves + 7) / 8;      // 8 waves (256 threads) per block
    k_row_stats<<<blocks1, 256, 0, stream>>>(logits, lse, pred, BT);
    k_ctc<<<B, 128, 0, stream>>>(logits, lse, labels, label_len, logit_len, loss, T, C, L);
    k_edit<<<(B + 255) / 256, 256, 0, stream>>>(pred, labels, label_len, logit_len, ed, B, T, L);
    k_reduce<<<1, 256, 0, stream>>>(loss, ed, (float*)d_out, B);
}